// ComplexDilatedConv_86199993631289
// MI455X (gfx1250) — compile-verified
//
#include <hip/hip_runtime.h>

typedef float v2f __attribute__((ext_vector_type(2)));
typedef float v8f __attribute__((ext_vector_type(8)));

constexpr int B_   = 8;
constexpr int CIN  = 64;
constexpr int T_   = 16384;
constexpr int MID  = 128;
constexpr int KW   = 3;
constexpr float EPSV = 1e-5f;

constexpr int TILE = 64;   // output time steps per workgroup
constexpr int YC   = 80;   // y-tile columns: t = t0 - 8 + c, c in [0,80)
constexpr int Z1C  = 68;   // z1 columns: c in [6,74)

static __device__ __forceinline__ v8f wmma4(v2f a, v2f b, v8f c) {
  // D = A(16x4 f32) * B(4x16 f32) + C(16x16 f32)
  return __builtin_amdgcn_wmma_f32_16x16x4_f32(
      /*neg_a=*/false, a, /*neg_b=*/false, b,
      /*c_mod=*/(short)0, c, /*reuse_a=*/false, /*reuse_b=*/false);
}

__global__ __launch_bounds__(256)
void cdc_fused(const float* __restrict__ x_re,  const float* __restrict__ x_im,
               const float* __restrict__ w_in_re, const float* __restrict__ w_in_im,
               const float* __restrict__ b_in_re, const float* __restrict__ b_in_im,
               const float* __restrict__ alpha_in_p,
               const float* __restrict__ ln_in_w, const float* __restrict__ ln_in_b,
               const float* __restrict__ dw_re,  const float* __restrict__ dw_im,
               const float* __restrict__ db_re,  const float* __restrict__ db_im,
               const float* __restrict__ alpha_out_p,
               const float* __restrict__ ln_out_w, const float* __restrict__ ln_out_b,
               const float* __restrict__ w_out_re, const float* __restrict__ w_out_im,
               const float* __restrict__ b_out_re, const float* __restrict__ b_out_im,
               float* __restrict__ out)
{
  extern __shared__ float sm[];
  float* xs_re = sm;                    // CIN * YC
  float* xs_im = xs_re + CIN * YC;
  float* yb_re = xs_im + CIN * YC;      // MID * YC
  float* yb_im = yb_re + MID * YC;
  float* z1_re = yb_im + MID * YC;      // MID * Z1C
  float* z1_im = z1_re + MID * Z1C;

  const int tid   = threadIdx.x;
  const int lane  = tid & 31;
  const int wv    = tid >> 5;
  const int lrow  = lane & 15;
  const int lhalf = lane >> 4;
  const int b     = blockIdx.y;
  const int t0    = blockIdx.x * TILE;

  const float alpha_in  = alpha_in_p[0];
  const float alpha_out = alpha_out_p[0];

  // ---------------- load x tile (+halo, zero padded) into LDS ----------------
  for (int idx = tid; idx < CIN * YC; idx += 256) {
    const int cin = idx / YC;
    const int c   = idx % YC;
    const int t   = t0 - 8 + c;
    const bool ok = (t >= 0) && (t < T_);
    const size_t g = (size_t)(b * CIN + cin) * T_ + (ok ? t : 0);
    xs_re[idx] = ok ? x_re[g] : 0.f;
    xs_im[idx] = ok ? x_im[g] : 0.f;
  }
  __syncthreads();

  // ---------------- stage A: complex pointwise-in via f32 WMMA ----------------
  {
    const int row = 16 * wv + lrow;   // MID output channel of this wave's M-tile
    v2f ar[16], ai[16];
#pragma unroll
    for (int k = 0; k < 16; ++k) {
      const int kb = 4 * k + 2 * lhalf;
      v2f tr = { w_in_re[row * CIN + kb], w_in_re[row * CIN + kb + 1] };
      v2f ti = { w_in_im[row * CIN + kb], w_in_im[row * CIN + kb + 1] };
      ar[k] = tr; ai[k] = ti;
    }
    for (int nt = 0; nt < 5; ++nt) {
      const int c = 16 * nt + lrow;
      v8f accR = {}, accN = {}, accI = {};
#pragma unroll
      for (int k = 0; k < 16; ++k) {
        const int kb = 4 * k + 2 * lhalf;
        v2f br = { xs_re[kb * YC + c], xs_re[(kb + 1) * YC + c] };
        v2f bi = { xs_im[kb * YC + c], xs_im[(kb + 1) * YC + c] };
        accR = wmma4(ar[k], br, accR);   // Wr*Xr
        accN = wmma4(ai[k], bi, accN);   // Wi*Xi  (subtract later)
        accI = wmma4(ar[k], bi, accI);   // Wr*Xi
        accI = wmma4(ai[k], br, accI);   // + Wi*Xr
      }
      const int t = t0 - 8 + c;
      const bool ok = (t >= 0) && (t < T_);
#pragma unroll
      for (int v = 0; v < 8; ++v) {
        const int ch = 16 * wv + v + 8 * lhalf;
        float yr = accR[v] - accN[v] + b_in_re[ch];
        float yi = accI[v] + b_in_im[ch];
        yr = yr >= 0.f ? yr : alpha_in * yr;
        yi = yi >= 0.f ? yi : alpha_in * yi;
        yb_re[ch * YC + c] = ok ? yr : 0.f;   // zero => correct SAME padding
        yb_im[ch * YC + c] = ok ? yi : 0.f;
      }
    }
  }
  __syncthreads();

  // ---------------- LN over channels (per column, re/im separately) ----------------
  if (tid < 2 * YC) {
    const int comp = tid / YC;
    const int c    = tid % YC;
    float* buf = comp ? yb_im : yb_re;
    float s = 0.f, s2 = 0.f;
    for (int ch = 0; ch < MID; ++ch) { float v = buf[ch * YC + c]; s += v; s2 += v * v; }
    const float mu  = s * (1.f / MID);
    const float var = s2 * (1.f / MID) - mu * mu;
    const float rs  = rsqrtf(var + EPSV);
    const int  t  = t0 - 8 + c;
    const bool ok = (t >= 0) && (t < T_);
    for (int ch = 0; ch < MID; ++ch) {
      const float v = (buf[ch * YC + c] - mu) * rs * ln_in_w[ch] + ln_in_b[ch];
      buf[ch * YC + c] = ok ? v : 0.f;
    }
  }
  __syncthreads();

  // ---------------- two complex dilated depthwise convs (VALU on LDS) ----------------
  {
    const int ch   = tid & 127;
    const int half = tid >> 7;
    float w1r[3], w1i[3], w2r[3], w2i[3];
#pragma unroll
    for (int j = 0; j < KW; ++j) {
      w1r[j] = dw_re[ch * KW + j];
      w1i[j] = dw_im[ch * KW + j];
      w2r[j] = dw_re[(MID + ch) * KW + j];
      w2i[j] = dw_im[(MID + ch) * KW + j];
    }
    const float b1r = db_re[ch],       b1i = db_im[ch];
    const float b2r = db_re[MID + ch], b2i = db_im[MID + ch];

    // conv #1: z1 on c in [6,74), taps at c-2, c, c+2 (dilation 2)
    for (int i = 0; i < 34; ++i) {
      const int c = 6 + half * 34 + i;
      const float yrm = yb_re[ch * YC + c - 2], yr0 = yb_re[ch * YC + c], yrp = yb_re[ch * YC + c + 2];
      const float yim = yb_im[ch * YC + c - 2], yi0 = yb_im[ch * YC + c], yip = yb_im[ch * YC + c + 2];
      const float zr = w1r[0]*yrm + w1r[1]*yr0 + w1r[2]*yrp
                     - (w1i[0]*yim + w1i[1]*yi0 + w1i[2]*yip) + b1r;
      const float zi = w1r[0]*yim + w1r[1]*yi0 + w1r[2]*yip
                     + (w1i[0]*yrm + w1i[1]*yr0 + w1i[2]*yrp) + b1i;
      const int  t  = t0 - 8 + c;
      const bool ok = (t >= 0) && (t < T_);
      z1_re[ch * Z1C + (c - 6)] = ok ? zr : 0.f;
      z1_im[ch * Z1C + (c - 6)] = ok ? zi : 0.f;
    }
    __syncthreads();

    // conv #2 + PReLU(alpha_out): outputs c in [8,72) overwrite yb
    for (int i = 0; i < 32; ++i) {
      const int c  = 8 + half * 32 + i;
      const int zc = c - 6;
      const float arm = z1_re[ch * Z1C + zc - 2], ar0 = z1_re[ch * Z1C + zc], arp = z1_re[ch * Z1C + zc + 2];
      const float aim = z1_im[ch * Z1C + zc - 2], ai0 = z1_im[ch * Z1C + zc], aip = z1_im[ch * Z1C + zc + 2];
      float zr = w2r[0]*arm + w2r[1]*ar0 + w2r[2]*arp
               - (w2i[0]*aim + w2i[1]*ai0 + w2i[2]*aip) + b2r;
      float zi = w2r[0]*aim + w2r[1]*ai0 + w2r[2]*aip
               + (w2i[0]*arm + w2i[1]*ar0 + w2i[2]*arp) + b2i;
      zr = zr >= 0.f ? zr : alpha_out * zr;
      zi = zi >= 0.f ? zi : alpha_out * zi;
      yb_re[ch * YC + c] = zr;
      yb_im[ch * YC + c] = zi;
    }
  }
  __syncthreads();

  // ---------------- output LN over channels ----------------
  if (tid < 128) {
    const int comp = tid >> 6;
    const int c    = 8 + (tid & 63);
    float* buf = comp ? yb_im : yb_re;
    float s = 0.f, s2 = 0.f;
    for (int ch = 0; ch < MID; ++ch) { float v = buf[ch * YC + c]; s += v; s2 += v * v; }
    const float mu  = s * (1.f / MID);
    const float var = s2 * (1.f / MID) - mu * mu;
    const float rs  = rsqrtf(var + EPSV);
    for (int ch = 0; ch < MID; ++ch)
      buf[ch * YC + c] = (buf[ch * YC + c] - mu) * rs * ln_out_w[ch] + ln_out_b[ch];
  }
  __syncthreads();

  // ---------------- stage C: complex pointwise-out + residual via f32 WMMA ----------------
  {
    const int mt  = wv & 3;
    const int row = 16 * mt + lrow;        // output channel in [0,64)
    const size_t outIm = (size_t)B_ * CIN * T_;
#pragma unroll
    for (int j = 0; j < 2; ++j) {
      const int nt = (wv >> 2) + 2 * j;    // 4x4 (M,N) tile jobs over 8 waves
      const int c  = 8 + 16 * nt + lrow;
      v8f accR = {}, accN = {}, accI = {};
#pragma unroll 8
      for (int k = 0; k < 32; ++k) {
        const int kb = 4 * k + 2 * lhalf;
        v2f a_r = { w_out_re[row * MID + kb], w_out_re[row * MID + kb + 1] };
        v2f a_i = { w_out_im[row * MID + kb], w_out_im[row * MID + kb + 1] };
        v2f b_r = { yb_re[kb * YC + c], yb_re[(kb + 1) * YC + c] };
        v2f b_i = { yb_im[kb * YC + c], yb_im[(kb + 1) * YC + c] };
        accR = wmma4(a_r, b_r, accR);
        accN = wmma4(a_i, b_i, accN);
        accI = wmma4(a_r, b_i, accI);
        accI = wmma4(a_i, b_r, accI);
      }
      const int t = t0 + 16 * nt + lrow;
#pragma unroll
      for (int v = 0; v < 8; ++v) {
        const int ch = 16 * mt + v + 8 * lhalf;
        const float orr = accR[v] - accN[v] + b_out_re[ch] + xs_re[ch * YC + c];
        const float oii = accI[v]            + b_out_im[ch] + xs_im[ch * YC + c];
        const size_t g = (size_t)(b * CIN + ch) * T_ + t;
        out[g]         = orr;
        out[outIm + g] = oii;
      }
    }
  }
}

extern "C" void kernel_launch(void* const* d_in, const int* in_sizes, int n_in,
                              void* d_out, int out_size, void* d_ws, size_t ws_size,
                              hipStream_t stream) {
  (void)in_sizes; (void)n_in; (void)out_size; (void)d_ws; (void)ws_size;
  const float* p[20];
  for (int i = 0; i < 20; ++i) p[i] = (const float*)d_in[i];
  dim3 grid(T_ / TILE, B_);
  dim3 block(256);
  const size_t shmem =
      (size_t)(2 * CIN * YC + 2 * MID * YC + 2 * MID * Z1C) * sizeof(float);
  cdc_fused<<<grid, block, shmem, stream>>>(
      p[0], p[1], p[2], p[3], p[4], p[5], p[6], p[7], p[8], p[9],
      p[10], p[11], p[12], p[13], p[14], p[15], p[16], p[17], p[18], p[19],
      (float*)d_out);
}